// SGC_57509612093517
// MI455X (gfx1250) — compile-verified
//
#include <hip/hip_runtime.h>

typedef float v2f __attribute__((ext_vector_type(2)));
typedef float v8f __attribute__((ext_vector_type(8)));

#define TB 256
#define DIM 128

// ---------------- degree / normalization ----------------

__global__ void k_deg_init(float* __restrict__ deg, int n) {
    int i = blockIdx.x * blockDim.x + threadIdx.x;
    if (i < n) deg[i] = 1.0f;   // self-loop contributes 1 to every target
}

__global__ void k_deg_count(const long long* __restrict__ ei,
                            float* __restrict__ deg, int e) {
    int i = blockIdx.x * blockDim.x + threadIdx.x;
    if (i < e) {
        int c = (int)ei[(size_t)e + i];   // col = edge_index[1][i]
        atomicAdd(&deg[c], 1.0f);
    }
}

__global__ void k_rsqrt_inplace(float* __restrict__ deg, int n) {
    int i = blockIdx.x * blockDim.x + threadIdx.x;
    if (i < n) {
        float d = deg[i];
        deg[i] = (d > 0.0f) ? rsqrtf(d) : 0.0f;
    }
}

// ---------------- one propagation hop: y = A_hat @ x ----------------

// self-loop term: y[i,:] = dinv[i]^2 * x[i,:]   (fully initializes y)
__global__ void k_hop_init(const float* __restrict__ x,
                           const float* __restrict__ dinv,
                           float* __restrict__ y, int n) {
    long i = (long)blockIdx.x * blockDim.x + threadIdx.x;   // one float4 each
    long total = (long)n * (DIM / 4);
    if (i < total) {
        int node = (int)(i >> 5);           // DIM/4 == 32 quads per node
        float s = dinv[node];
        s *= s;
        float4 v = ((const float4*)x)[i];
        v.x *= s; v.y *= s; v.z *= s; v.w *= s;
        ((float4*)y)[i] = v;
    }
}

// edge term: y[c,:] += dinv[r]*dinv[c] * x[r,:]; 32 lanes per edge, float4/lane
__global__ void k_hop_edges(const long long* __restrict__ ei,
                            const float* __restrict__ dinv,
                            const float* __restrict__ x,
                            float* __restrict__ y, int e) {
    long t = (long)blockIdx.x * blockDim.x + threadIdx.x;
    int edge = (int)(t >> 5);
    int lane = (int)(t & 31);
    if (edge < e) {
        int r = (int)ei[edge];
        int c = (int)ei[(size_t)e + edge];
        float coef = dinv[r] * dinv[c];
        float4 v = ((const float4*)(x + (size_t)r * DIM))[lane];
        float* yp = y + (size_t)c * DIM + lane * 4;
        atomicAdd(yp + 0, coef * v.x);
        atomicAdd(yp + 1, coef * v.y);
        atomicAdd(yp + 2, coef * v.z);
        atomicAdd(yp + 3, coef * v.w);
    }
}

// ---------------- dense projection: out = relu(X @ W^T + b) ----------------
// fp32 WMMA 16x16x4. One wave -> 16 rows x 128 cols (8 x v8f accumulators).
// W (128x128, row-major [H,D]) staged in LDS once per block.

__global__ void __launch_bounds__(TB) k_gemm_relu(const float* __restrict__ X,
                                                  const float* __restrict__ W,
                                                  const float* __restrict__ bias,
                                                  float* __restrict__ Out, int n) {
    __shared__ float Ws[DIM * DIM];   // 64 KB of the 320 KB WGP LDS
    for (int i = threadIdx.x; i < DIM * DIM; i += TB) Ws[i] = W[i];
    __syncthreads();

    const int lane = threadIdx.x & 31;
    const int wave = threadIdx.x >> 5;
    const int rowBase = (blockIdx.x * (TB / 32) + wave) * 16;
    if (rowBase + 16 > n) return;     // wave-uniform: EXEC stays all-1s

    const int half = lane >> 4;       // 0: K pair (k,k+1); 1: (k+2,k+3)
    const int l16  = lane & 15;

    v8f acc[8] = {};                  // 8 N-tiles of 16 cols -> 128 cols

    // A row pointer for this lane (ISA 32-bit A 16x4 layout)
    const float* arow = X + (size_t)(rowBase + l16) * DIM + 2 * half;

    for (int k = 0; k < DIM; k += 4) {
        v2f a = *(const v2f*)(arow + k);           // global_load_b64
#pragma unroll
        for (int t = 0; t < 8; ++t) {
            // B[kk][nn] = W^T[k+kk][nb+nn] = W[nb+nn][k+kk]; same lane pattern as A
            const float* bp = Ws + (t * 16 + l16) * DIM + k + 2 * half;
            v2f b = *(const v2f*)bp;               // ds_load_b64
            acc[t] = __builtin_amdgcn_wmma_f32_16x16x4_f32(
                false, a, false, b, (short)0, acc[t], false, false);
        }
    }

    // C/D layout: VGPR j -> row (j + 8*half), col l16 within the 16x16 tile
#pragma unroll
    for (int t = 0; t < 8; ++t) {
        const int nb = t * 16;
        const float bb = bias[nb + l16];
#pragma unroll
        for (int j = 0; j < 8; ++j) {
            const int m = rowBase + j + 8 * half;
            float v = acc[t][j] + bb;
            Out[(size_t)m * DIM + nb + l16] = (v > 0.0f) ? v : 0.0f;
        }
    }
}

// ---------------- driver ----------------

extern "C" void kernel_launch(void* const* d_in, const int* in_sizes, int n_in,
                              void* d_out, int out_size, void* d_ws, size_t ws_size,
                              hipStream_t stream) {
    const float*     x  = (const float*)d_in[0];
    const long long* ei = (const long long*)d_in[1];   // int64 [2,E]
    const float*     W  = (const float*)d_in[2];       // [H,D]
    const float*     b  = (const float*)d_in[3];       // [H]
    float* out = (float*)d_out;

    const int n = in_sizes[0] / DIM;   // 50000
    const int e = in_sizes[1] / 2;     // 600000

    // workspace: dinv (n floats, 64K-float slot) then hop buffer y1 (n*128)
    float* dinv = (float*)d_ws;
    float* y1   = dinv + (1 << 16);

    // normalization
    k_deg_init<<<(n + TB - 1) / TB, TB, 0, stream>>>(dinv, n);
    k_deg_count<<<(e + TB - 1) / TB, TB, 0, stream>>>(ei, dinv, e);
    k_rsqrt_inplace<<<(n + TB - 1) / TB, TB, 0, stream>>>(dinv, n);

    const long quads = (long)n * (DIM / 4);
    const long elth  = (long)e * 32;

    // hop 1: x -> y1
    k_hop_init <<<(int)((quads + TB - 1) / TB), TB, 0, stream>>>(x, dinv, y1, n);
    k_hop_edges<<<(int)((elth  + TB - 1) / TB), TB, 0, stream>>>(ei, dinv, x, y1, e);

    // hop 2: y1 -> out
    k_hop_init <<<(int)((quads + TB - 1) / TB), TB, 0, stream>>>(y1, dinv, out, n);
    k_hop_edges<<<(int)((elth  + TB - 1) / TB), TB, 0, stream>>>(ei, dinv, y1, out, e);

    // projection, in place on out (each wave reads its 16 rows before storing them)
    const int waves  = n / 16;                       // 3125 (n is a multiple of 16)
    const int blocks = (waves + (TB / 32) - 1) / (TB / 32);
    k_gemm_relu<<<blocks, TB, 0, stream>>>(out, W, b, out, n);
}